// KANC_MLP_28552942584480
// MI455X (gfx1250) — compile-verified
//
#include <hip/hip_runtime.h>
#include <hip/hip_bf16.h>

typedef float v2f __attribute__((ext_vector_type(2)));
typedef float v8f __attribute__((ext_vector_type(8)));

#define B_SZ   64
#define H_IN   128
#define W_IN   384      // 3 x 128 concatenated along W
#define HO_P   32       // pooled H (even conv rows only)
#define WO_P   96       // pooled W (max over pairs of 192)
#define KFLAT  15360    // 5 * 32 * 96
#define NHID   256

// ---------------------------------------------------------------------------
// Fetch from the virtual concatenation of x1|x2|x3 with zero padding.
// ---------------------------------------------------------------------------
__device__ __forceinline__ float fetch_x(const float* __restrict__ x1,
                                         const float* __restrict__ x2,
                                         const float* __restrict__ x3,
                                         int b, int h, int w) {
    if ((unsigned)h >= (unsigned)H_IN || (unsigned)w >= (unsigned)W_IN) return 0.0f;
    const float* src;
    int wl;
    if (w < 128)      { src = x1; wl = w; }
    else if (w < 256) { src = x2; wl = w - 128; }
    else              { src = x3; wl = w - 256; }
    return src[((size_t)b * H_IN + h) * 128 + wl];
}

// ---------------------------------------------------------------------------
// KAN evaluation of one 3x3 patch (cols coff..coff+2 of a 3x5 window):
// out[c] = sum_i silu(p_i)*base_w[c,i] + sum_{i,k} Bspline_k(p_i)*spline_w[c,i,k]
// Uniform grid g[t] = (t-3)*0.4 - 1, cubic Cox-de Boor, bases length 8.
// ---------------------------------------------------------------------------
__device__ __forceinline__ void kan_eval(const float win[3][5], int coff,
                                         const float* __restrict__ sbw,
                                         const float* __restrict__ ssw,
                                         float out[5]) {
#pragma unroll
    for (int c = 0; c < 5; ++c) out[c] = 0.0f;
#pragma unroll
    for (int kh = 0; kh < 3; ++kh) {
#pragma unroll
        for (int kw = 0; kw < 3; ++kw) {
            const int i = kh * 3 + kw;
            const float v = win[kh][coff + kw];
            const float sv = v / (1.0f + __expf(-v));       // SiLU
            float bs[11];
#pragma unroll
            for (int t = 0; t < 11; ++t) {
                const float gl = (float)(t - 3) * 0.4f - 1.0f;
                const float gr = (float)(t - 2) * 0.4f - 1.0f;
                bs[t] = (v >= gl && v < gr) ? 1.0f : 0.0f;
            }
#pragma unroll
            for (int k = 1; k <= 3; ++k) {
                const float inv = 1.0f / (0.4f * (float)k);
#pragma unroll
                for (int t = 0; t < 11; ++t) {
                    if (t + k < 11) {
                        const float gl = (float)(t - 3) * 0.4f - 1.0f;
                        const float gr = (float)(t + k - 2) * 0.4f - 1.0f;
                        bs[t] = ((v - gl) * bs[t] + (gr - v) * bs[t + 1]) * inv;
                    }
                }
            }
#pragma unroll
            for (int c = 0; c < 5; ++c) {
                float acc = sv * sbw[c * 9 + i];
                const float* sw = &ssw[(c * 9 + i) * 8];
#pragma unroll
                for (int kk = 0; kk < 8; ++kk) acc = fmaf(bs[kk], sw[kk], acc);
                out[c] += acc;
            }
        }
    }
}

// ---------------------------------------------------------------------------
// Kernel A: fused patch-extract + KAN + (1,2)/(2,2) max-pool.
// Only even conv rows are evaluated (pool H window=1 stride=2).
// One thread -> one pooled (b, ohp, owp): evaluates conv cols 2*owp, 2*owp+1.
// Writes pooled in (B, 5, 32, 96) order = GEMM K layout.
// ---------------------------------------------------------------------------
__global__ __launch_bounds__(256) void kan_conv_pool(
    const float* __restrict__ x1, const float* __restrict__ x2,
    const float* __restrict__ x3, const float* __restrict__ base_w,
    const float* __restrict__ spline_w, float* __restrict__ pooled) {
    __shared__ float s_bw[45];
    __shared__ float s_sw[360];
    const int t = threadIdx.x;
    for (int j = t; j < 45; j += 256)  s_bw[j] = base_w[j];
    for (int j = t; j < 360; j += 256) s_sw[j] = spline_w[j];
    __syncthreads();

    const int gid = blockIdx.x * 256 + t;          // exactly 64*32*96 = 196608
    const int owp = gid % WO_P;
    const int ohp = (gid / WO_P) % HO_P;
    const int b   = gid / (WO_P * HO_P);

    const int r0 = 4 * ohp - 1;                    // conv row 2*ohp -> input rows
    const int c0 = 4 * owp - 1;                    // conv cols 2*owp,2*owp+1 -> 5 input cols
    float win[3][5];
#pragma unroll
    for (int rr = 0; rr < 3; ++rr)
#pragma unroll
        for (int cc = 0; cc < 5; ++cc)
            win[rr][cc] = fetch_x(x1, x2, x3, b, r0 + rr, c0 + cc);

    float o0[5], o1[5];
    kan_eval(win, 0, s_bw, s_sw, o0);
    kan_eval(win, 2, s_bw, s_sw, o1);

#pragma unroll
    for (int c = 0; c < 5; ++c) {
        const float m = fmaxf(o0[c], o1[c]);
        pooled[((size_t)b * 5 + c) * (HO_P * WO_P) + ohp * WO_P + owp] = m;
    }
}

// ---------------------------------------------------------------------------
// Kernel B: h = pooled(64 x 15360) @ w1^T(15360 x 256) + b1, via f32 WMMA.
// One 16x16 C tile per block; K split 4-ways across the block's 4 waves with a
// deterministic LDS reduction. A frag: lane m=lane&15, k-pair = 2*(lane>>4).
// B frag: B[k][n] = w1[n][k]  -> per-lane float2 at row n = nt*16 + (lane&15).
// ---------------------------------------------------------------------------
__global__ __launch_bounds__(128) void kan_gemm1(
    const float* __restrict__ pooled, const float* __restrict__ w1,
    const float* __restrict__ b1, float* __restrict__ h) {
    const int lane = threadIdx.x & 31;
    const int wave = threadIdx.x >> 5;             // 0..3 -> K chunk
    const int mt = blockIdx.x & 3;                 // 64/16 M tiles
    const int nt = blockIdx.x >> 2;                // 256/16 N tiles
    const int row = lane & 15;
    const int hi  = lane >> 4;

    const float* Abase = pooled + (size_t)(mt * 16 + row) * KFLAT + 2 * hi;
    const float* Bbase = w1     + (size_t)(nt * 16 + row) * KFLAT + 2 * hi;

    v8f acc = {};
    const int kchunk = KFLAT / 4;                  // 3840
    const int k0 = wave * kchunk;
    const int k1 = k0 + kchunk;
    for (int k = k0; k < k1; k += 4) {
        v2f a = *(const v2f*)(Abase + k);
        v2f b = *(const v2f*)(Bbase + k);
        acc = __builtin_amdgcn_wmma_f32_16x16x4_f32(
            /*neg_a=*/false, a, /*neg_b=*/false, b,
            /*c_mod=*/(short)0, acc, /*reuse_a=*/false, /*reuse_b=*/false);
    }

    __shared__ float red[4 * 32 * 8];              // 4 KB
#pragma unroll
    for (int r = 0; r < 8; ++r) red[wave * 256 + lane * 8 + r] = acc[r];
    __syncthreads();
    if (wave == 0) {
#pragma unroll
        for (int r = 0; r < 8; ++r) {
            const int idx = lane * 8 + r;
            const float s = red[idx] + red[256 + idx] + red[512 + idx] + red[768 + idx];
            const int m = mt * 16 + r + 8 * hi;    // C layout: VGPR r, lane half
            const int n = nt * 16 + row;
            h[m * NHID + n] = s + b1[n];
        }
    }
}

// ---------------------------------------------------------------------------
// Kernel C: logits = h @ w2^T + b2, then log_softmax over 2 classes.
// ---------------------------------------------------------------------------
__global__ __launch_bounds__(64) void kan_head(
    const float* __restrict__ h, const float* __restrict__ w2,
    const float* __restrict__ b2, float* __restrict__ out) {
    const int b = threadIdx.x;                     // 64 batches
    float l0 = b2[0], l1 = b2[1];
    for (int j = 0; j < NHID; ++j) {
        const float v = h[b * NHID + j];
        l0 = fmaf(v, w2[j], l0);
        l1 = fmaf(v, w2[NHID + j], l1);
    }
    const float m = fmaxf(l0, l1);
    const float lse = m + __logf(__expf(l0 - m) + __expf(l1 - m));
    out[b * 2 + 0] = l0 - lse;
    out[b * 2 + 1] = l1 - lse;
}

// ---------------------------------------------------------------------------
extern "C" void kernel_launch(void* const* d_in, const int* in_sizes, int n_in,
                              void* d_out, int out_size, void* d_ws, size_t ws_size,
                              hipStream_t stream) {
    const float* x1       = (const float*)d_in[0];
    const float* x2       = (const float*)d_in[1];
    const float* x3       = (const float*)d_in[2];
    const float* base_w   = (const float*)d_in[3];
    const float* spline_w = (const float*)d_in[4];
    // d_in[5] = grid: uniform, reconstructed in-kernel from its closed form
    const float* w1       = (const float*)d_in[6];
    const float* b1       = (const float*)d_in[7];
    const float* w2       = (const float*)d_in[8];
    const float* b2       = (const float*)d_in[9];

    float* pooled = (float*)d_ws;                                   // 64*15360 f32
    float* h      = (float*)((char*)d_ws + (size_t)B_SZ * KFLAT * 4); // 64*256 f32
    float* out    = (float*)d_out;

    kan_conv_pool<<<768, 256, 0, stream>>>(x1, x2, x3, base_w, spline_w, pooled);
    kan_gemm1<<<64, 128, 0, stream>>>(pooled, w1, b1, h);
    kan_head<<<1, 64, 0, stream>>>(h, w2, b2, out);
}